// SobelOperator_88828513616269
// MI455X (gfx1250) — compile-verified
//
#include <hip/hip_runtime.h>
#include <math.h>

// Sobel magnitude: x [B,1,512,512] f32 -> sqrt(gx^2 + gy^2 + 1e-4), padding=1 (zeros).
// Memory-bound (AI ~3.75 flop/B): optimize for the 23.3 TB/s HBM path using the
// CDNA5 async global->LDS copy engine + b128 traffic, not WMMA.

#define IW 512
#define IH 512
#define TW 128           // output tile width  (x)
#define TH 32            // output tile height (y)
#define LW 136           // LDS row width in floats: covers [x0-4, x0+132), 16B multiple
#define LH 34            // LDS rows: covers [y0-1, y0+TH+1)
#define NTHREADS 256     // 8 wave32s

typedef __attribute__((address_space(3))) const float lds_cfloat;

union F4 { float4 v; float f[4]; };

__global__ __launch_bounds__(NTHREADS)
void sobel_async_tile_kernel(const float* __restrict__ x, float* __restrict__ out) {
    __shared__ float tile[LH][LW];   // 34*136*4 = 18,496 B of the 320 KB/WGP LDS

    const int tid   = threadIdx.x;
    const int x0    = blockIdx.x * TW;
    const int y0    = blockIdx.y * TH;
    const size_t plane = (size_t)blockIdx.z * (size_t)(IW * IH);
    const float* src = x + plane;

    // ---------------------------------------------------------------------
    // Stage the haloed input tile into LDS with async global->LDS b128 DMA.
    // LDS col 0 maps to gx = x0-4; since x0 % 128 == 0 and IW % 4 == 0,
    // every 4-float chunk is either fully in-bounds or fully out-of-bounds.
    // OOB chunks are zero-filled (padding=1 semantics) with plain ds stores.
    // ---------------------------------------------------------------------
    const int nchunk = LH * (LW / 4);            // 34 * 34 = 1156 chunks of 16 B
    for (int i = tid; i < nchunk; i += NTHREADS) {
        const int r  = i / (LW / 4);
        const int c4 = (i % (LW / 4)) * 4;       // float column within LDS row
        const int gy = y0 - 1 + r;
        const int gx = x0 - 4 + c4;
        if (gy >= 0 && gy < IH && gx >= 0 && (gx + 4) <= IW) {
            const unsigned lds_b =
                (unsigned)(unsigned long long)(lds_cfloat*)&tile[r][c4];
            const unsigned long long gaddr =
                (unsigned long long)(const void*)(src + (size_t)gy * IW + gx);
            // CDNA5 async copy: 16 B per enabled lane, global -> LDS, ASYNCcnt.
            asm volatile("global_load_async_to_lds_b128 %0, %1, off"
                         :: "v"(lds_b), "v"(gaddr)
                         : "memory");
        } else {
            *(float4*)&tile[r][c4] = make_float4(0.f, 0.f, 0.f, 0.f);
        }
    }
    // Drain this wave's async-DMA ops, then make LDS visible workgroup-wide.
    asm volatile("s_wait_asynccnt 0x0" ::: "memory");
    __syncthreads();

    // ---------------------------------------------------------------------
    // Compute: each thread produces a 4x4 output patch from a 6x6 LDS window.
    // 256 threads = 32 x-strips (4 wide) x 8 y-strips (4 tall) = 128x32 tile.
    // ---------------------------------------------------------------------
    const int xl0 = (tid & 31) * 4;              // output-local x of this strip
    const int yl0 = (tid >> 5) * 4;              // output-local y of this strip

    // Output (yl, xl) center lives at LDS [yl+1][xl+4]; window spans
    // rows yl0 .. yl0+5, cols xl0+3 .. xl0+8.
    float s[6][6];
#pragma unroll
    for (int r = 0; r < 6; ++r)
#pragma unroll
        for (int c = 0; c < 6; ++c)
            s[r][c] = tile[yl0 + r][xl0 + 3 + c];

    float* dst = out + plane;
#pragma unroll
    for (int i = 0; i < 4; ++i) {
        F4 o;
#pragma unroll
        for (int j = 0; j < 4; ++j) {
            // Cross-correlation with kx = [[1,0,-1],[2,0,-2],[1,0,-1]]/4
            const float gxv = (        s[i    ][j] - s[i    ][j + 2]
                               + 2.f * (s[i + 1][j] - s[i + 1][j + 2])
                               +        s[i + 2][j] - s[i + 2][j + 2]) * 0.25f;
            // ky = [[1,2,1],[0,0,0],[-1,-2,-1]]/4
            const float gyv = (  s[i    ][j] + 2.f * s[i    ][j + 1] + s[i    ][j + 2]
                               - s[i + 2][j] - 2.f * s[i + 2][j + 1] - s[i + 2][j + 2]) * 0.25f;
            o.f[j] = sqrtf(gxv * gxv + gyv * gyv + 1e-4f);
        }
        // 16B-aligned b128 store: x0+xl0 is a multiple of 4 floats.
        *(float4*)&dst[(size_t)(y0 + yl0 + i) * IW + (x0 + xl0)] = o.v;
    }
}

extern "C" void kernel_launch(void* const* d_in, const int* in_sizes, int n_in,
                              void* d_out, int out_size, void* d_ws, size_t ws_size,
                              hipStream_t stream) {
    const float* xin = (const float*)d_in[0];
    float* out = (float*)d_out;
    const int batch = in_sizes[0] / (IW * IH);   // 64 for the reference shapes
    dim3 grid(IW / TW, IH / TH, batch);          // 4 x 16 x 64 = 4096 blocks
    sobel_async_tile_kernel<<<grid, NTHREADS, 0, stream>>>(xin, out);
}